// fCGModule_4698694222683
// MI455X (gfx1250) — compile-verified
//
#include <hip/hip_runtime.h>
#include <math.h>

typedef __attribute__((ext_vector_type(2))) float v2f;
typedef __attribute__((ext_vector_type(8))) float v8f;

#define MAXL 5
#define BATCH 256

// ---------------------------------------------------------------------------
// Compile-time enumeration of the CG triples in the exact reference order.
// ---------------------------------------------------------------------------
struct Meta {
    int ntrip;
    int trip_l1[80], trip_l2[80], trip_l[80], trip_cgoff[80];
    int cg_total;
    int nfrag[MAXL + 1];
    int frag_l1[MAXL + 1][16], frag_l2[MAXL + 1][16], frag_cgoff[MAXL + 1][16];
    long long out_base[MAXL + 1];
    long long out_elems;
};

constexpr Meta build_meta() {
    Meta m{};
    int cg = 0;
    for (int l1 = 0; l1 <= MAXL; ++l1)
        for (int l2 = 0; l2 <= l1; ++l2) {
            int lmax = (l1 + l2 < MAXL) ? (l1 + l2) : MAXL;
            for (int l = l1 - l2; l <= lmax; ++l) {
                int t = m.ntrip++;
                m.trip_l1[t] = l1; m.trip_l2[t] = l2; m.trip_l[t] = l;
                m.trip_cgoff[t] = cg;
                int f = m.nfrag[l]++;
                m.frag_l1[l][f] = l1; m.frag_l2[l][f] = l2; m.frag_cgoff[l][f] = cg;
                cg += (2 * l1 + 1) * (2 * l2 + 1);
            }
        }
    m.cg_total = cg;
    long long off = 0;
    for (int l = 0; l <= MAXL; ++l) {
        m.out_base[l] = off;
        off += (long long)BATCH * (m.nfrag[l] * 256LL) * (2 * l + 1) * 2;
    }
    m.out_elems = off;
    return m;
}

constexpr Meta META = build_meta();
static_assert(META.ntrip == 69, "triple count");
static_assert(META.cg_total == 3587, "cg table size");

__constant__ Meta d_meta = META;

// ---------------------------------------------------------------------------
// Prep kernel: recompute CG tables into d_ws (deterministic each call).
// Layout: per triple t at trip_cgoff[t], (2l1+1)x(2l2+1) floats, entry [p][q]
// holds <l1 m1; l2 m2 | l (m1+m2)> with m1=p-l1, m2=q-l2 (0 if |m1+m2|>l).
// ---------------------------------------------------------------------------
__device__ inline double dfact(int n) {
    double r = 1.0;
    for (int i = 2; i <= n; ++i) r *= (double)i;
    return r;
}

__device__ float cg_coef(int l1, int l2, int l, int m1, int m2) {
    int m = m1 + m2;
    if (m < -l || m > l) return 0.0f;
    double pref = (double)(2 * l + 1) * dfact(l + l1 - l2) * dfact(l - l1 + l2) *
                  dfact(l1 + l2 - l) / dfact(l1 + l2 + l + 1);
    pref *= dfact(l + m) * dfact(l - m) * dfact(l1 - m1) * dfact(l1 + m1) *
            dfact(l2 - m2) * dfact(l2 + m2);
    int kmin = 0;
    if (-(l - l2 + m1) > kmin) kmin = -(l - l2 + m1);
    if (-(l - l1 - m2) > kmin) kmin = -(l - l1 - m2);
    int kmax = l1 + l2 - l;
    if (l1 - m1 < kmax) kmax = l1 - m1;
    if (l2 + m2 < kmax) kmax = l2 + m2;
    double s = 0.0;
    for (int k = kmin; k <= kmax; ++k) {
        double d = dfact(k) * dfact(l1 + l2 - l - k) * dfact(l1 - m1 - k) *
                   dfact(l2 + m2 - k) * dfact(l - l2 + m1 + k) * dfact(l - l1 - m2 + k);
        s += ((k & 1) ? -1.0 : 1.0) / d;
    }
    return (float)(sqrt(pref) * s);
}

__global__ void cg_prep_kernel(float* __restrict__ cg) {
    int idx = blockIdx.x * blockDim.x + threadIdx.x;
    if (idx >= d_meta.cg_total) return;
    int t = 0;
    for (int s = 0; s < d_meta.ntrip; ++s) {
        int n = (2 * d_meta.trip_l1[s] + 1) * (2 * d_meta.trip_l2[s] + 1);
        if (idx >= d_meta.trip_cgoff[s] && idx < d_meta.trip_cgoff[s] + n) t = s;
    }
    int l1 = d_meta.trip_l1[t], l2 = d_meta.trip_l2[t], l = d_meta.trip_l[t];
    int rel = idx - d_meta.trip_cgoff[t];
    int n2 = 2 * l2 + 1;
    int p = rel / n2, q = rel % n2;
    cg[idx] = cg_coef(l1, l2, l, p - l1, q - l2);
}

// ---------------------------------------------------------------------------
// Main kernel: one wave32 per (fragment f, batch b). Computes the full
// 16x16 x (2L+1) complex tile with V_WMMA_F32_16X16X4_F32.
//   A[i][k] = c_k * x1[b,i,p_k]   (CG folded into A; pad slots get c=0)
//   B[k][j] = x2[b,j,q_k]
//   D = A*B accumulated per output m; re = XX - YY, im = XY + YX.
// Lane layouts (ISA 7.12.2, 32-bit operands):
//   A 16x4 : lanes 0-15 hold K=0,1 ; lanes 16-31 hold K=2,3 (rows M=lane%16)
//   B 4x16 : lanes 0-15 hold K=0,1 ; lanes 16-31 hold K=2,3 (cols N=lane%16)
//   C/D    : VGPR r -> row r (lanes 0-15) / row r+8 (lanes 16-31), col lane%16
// Padded K-slots: m1 is CLAMPED into [m1lo,m1hi] so every LDS index is valid
// (unconditional ds_load); only the CG coefficient is zeroed via cndmask,
// which zeroes the A operand and hence the padded contribution.
// ---------------------------------------------------------------------------
template <int L>
__global__ void __launch_bounds__(32) cg_wmma_kernel(
    const float* __restrict__ a0, const float* __restrict__ a1,
    const float* __restrict__ a2, const float* __restrict__ a3,
    const float* __restrict__ a4, const float* __restrict__ a5,
    const float* __restrict__ cgbuf, float* __restrict__ outl)
{
    constexpr int NM = 2 * L + 1;
    constexpr long long CL = (long long)META.nfrag[L] * 256;  // channels of out_l

    const int f = blockIdx.x;
    const int b = blockIdx.y;
    const int lane = threadIdx.x;
    const int half = lane >> 4;     // K-slot group / output row group
    const int rc = lane & 15;       // A row i  /  B col j  /  D col

    const int l1 = d_meta.frag_l1[L][f];
    const int l2 = d_meta.frag_l2[L][f];
    const int n1 = 2 * l1 + 1;
    const int n2 = 2 * l2 + 1;
    const float* cg = cgbuf + d_meta.frag_cgoff[L][f];

    const float* in1;
    switch (l1) { case 0: in1 = a0; break; case 1: in1 = a1; break;
                  case 2: in1 = a2; break; case 3: in1 = a3; break;
                  case 4: in1 = a4; break; default: in1 = a5; }
    const float* in2;
    switch (l2) { case 0: in2 = a0; break; case 1: in2 = a1; break;
                  case 2: in2 = a2; break; case 3: in2 = a3; break;
                  case 4: in2 = a4; break; default: in2 = a5; }

    // Stage this batch's activations (complex de-interleaved) + CG into LDS.
    __shared__ float sx1[16 * 11], sy1[16 * 11];
    __shared__ float sx2[16 * 11], sy2[16 * 11];
    __shared__ float scg[121];

    const float2* p1 = (const float2*)(in1 + (size_t)b * 16 * n1 * 2);
    const float2* p2 = (const float2*)(in2 + (size_t)b * 16 * n2 * 2);
    for (int t = lane; t < 16 * n1; t += 32) { float2 v = p1[t]; sx1[t] = v.x; sy1[t] = v.y; }
    for (int t = lane; t < 16 * n2; t += 32) { float2 v = p2[t]; sx2[t] = v.x; sy2[t] = v.y; }
    for (int t = lane; t < n1 * n2; t += 32) scg[t] = cg[t];
    __syncthreads();

    const int row1 = rc * n1;   // base into sx1/sy1 for this lane's A row
    const int row2 = rc * n2;   // base into sx2/sy2 for this lane's B column

    v8f accRe[NM], accIm[NM];
#pragma unroll
    for (int mm = 0; mm < NM; ++mm) { accRe[mm] = (v8f)0.0f; accIm[mm] = (v8f)0.0f; }

#pragma unroll
    for (int mm = 0; mm < NM; ++mm) {
        const int m = mm - L;
        const int m1lo = (-l1 > m - l2) ? -l1 : (m - l2);
        const int m1hi = ( l1 < m + l2) ?  l1 : (m + l2);
        // Chunk the m1 diagonal into K=4 WMMA steps (uniform loop -> EXEC all 1s).
        for (int m1s = m1lo; m1s <= m1hi; m1s += 4) {
            const int mA = m1s + 2 * half;       // this lane's K-slots: 2h, 2h+1
            const int mB = mA + 1;
            const bool va = (mA <= m1hi);
            const bool vb = (mB <= m1hi);
            const int mAc = va ? mA : m1hi;      // clamp -> indices always valid
            const int mBc = vb ? mB : m1hi;
            const int pa = mAc + l1, qa = m - mAc + l2;
            const int pb = mBc + l1, qb = m - mBc + l2;

            const float ca = scg[pa * n2 + qa];  // unconditional LDS loads
            const float cb = scg[pb * n2 + qb];
            const float c0 = va ? ca : 0.0f;     // zero only the coefficient
            const float c1 = vb ? cb : 0.0f;

            const float x1a = sx1[row1 + pa], x1b = sx1[row1 + pb];
            const float y1a = sy1[row1 + pa], y1b = sy1[row1 + pb];
            v2f Ax  = {  c0 * x1a,  c1 * x1b };
            v2f Ay  = {  c0 * y1a,  c1 * y1b };
            v2f Ayn = { -c0 * y1a, -c1 * y1b };
            v2f Bx  = { sx2[row2 + qa], sx2[row2 + qb] };
            v2f By  = { sy2[row2 + qa], sy2[row2 + qb] };

            accRe[mm] = __builtin_amdgcn_wmma_f32_16x16x4_f32(
                false, Ax,  false, Bx, (short)0, accRe[mm], false, false);
            accRe[mm] = __builtin_amdgcn_wmma_f32_16x16x4_f32(
                false, Ayn, false, By, (short)0, accRe[mm], false, false);
            accIm[mm] = __builtin_amdgcn_wmma_f32_16x16x4_f32(
                false, Ax,  false, By, (short)0, accIm[mm], false, false);
            accIm[mm] = __builtin_amdgcn_wmma_f32_16x16x4_f32(
                false, Ay,  false, Bx, (short)0, accIm[mm], false, false);
        }
    }

    // Store: out_l[b, f*256 + i*16 + j, m, {re,im}] as contiguous float2 runs.
    float* outb = outl + ((size_t)b * CL + (size_t)f * 256) * NM * 2;
#pragma unroll
    for (int r = 0; r < 8; ++r) {
        const int i = r + 8 * half;
        float* dst = outb + (size_t)(i * 16 + rc) * NM * 2;
#pragma unroll
        for (int mm = 0; mm < NM; ++mm) {
            float2 v;
            v.x = accRe[mm][r];
            v.y = accIm[mm][r];
            *(float2*)(dst + mm * 2) = v;
        }
    }
}

// ---------------------------------------------------------------------------
// Launcher: prep CG tables into d_ws, then one launch per output l.
// ---------------------------------------------------------------------------
extern "C" void kernel_launch(void* const* d_in, const int* in_sizes, int n_in,
                              void* d_out, int out_size, void* d_ws, size_t ws_size,
                              hipStream_t stream) {
    const float* a0 = (const float*)d_in[0];
    const float* a1 = (const float*)d_in[1];
    const float* a2 = (const float*)d_in[2];
    const float* a3 = (const float*)d_in[3];
    const float* a4 = (const float*)d_in[4];
    const float* a5 = (const float*)d_in[5];
    float* cg  = (float*)d_ws;
    float* out = (float*)d_out;

    const int prep_blocks = (META.cg_total + 255) / 256;
    cg_prep_kernel<<<prep_blocks, 256, 0, stream>>>(cg);

    cg_wmma_kernel<0><<<dim3(META.nfrag[0], BATCH), 32, 0, stream>>>(
        a0, a1, a2, a3, a4, a5, cg, out + (size_t)META.out_base[0]);
    cg_wmma_kernel<1><<<dim3(META.nfrag[1], BATCH), 32, 0, stream>>>(
        a0, a1, a2, a3, a4, a5, cg, out + (size_t)META.out_base[1]);
    cg_wmma_kernel<2><<<dim3(META.nfrag[2], BATCH), 32, 0, stream>>>(
        a0, a1, a2, a3, a4, a5, cg, out + (size_t)META.out_base[2]);
    cg_wmma_kernel<3><<<dim3(META.nfrag[3], BATCH), 32, 0, stream>>>(
        a0, a1, a2, a3, a4, a5, cg, out + (size_t)META.out_base[3]);
    cg_wmma_kernel<4><<<dim3(META.nfrag[4], BATCH), 32, 0, stream>>>(
        a0, a1, a2, a3, a4, a5, cg, out + (size_t)META.out_base[4]);
    cg_wmma_kernel<5><<<dim3(META.nfrag[5], BATCH), 32, 0, stream>>>(
        a0, a1, a2, a3, a4, a5, cg, out + (size_t)META.out_base[5]);
}